// GraphConvolution_62672162783472
// MI455X (gfx1250) — compile-verified
//
#include <hip/hip_runtime.h>

typedef __attribute__((ext_vector_type(2))) float v2f;
typedef __attribute__((ext_vector_type(8))) float v8f;
typedef __attribute__((ext_vector_type(4))) unsigned int u32x4;
typedef __attribute__((ext_vector_type(8))) int i32x8;
typedef __attribute__((ext_vector_type(4))) int i32x4;

#define IN_DIM  256
#define OUT_DIM 128
#define LDS_PITCH 260   // 256 dwords/row + 4 dwords TDM pad -> stride%64 == 4 (bank-conflict-free A reads)

// ---------------------------------------------------------------------------
// Kernel 1: zero the output accumulator (float4-vectorized grid-stride).
// ---------------------------------------------------------------------------
__global__ void gcn_zero(float4* __restrict__ out, size_t n4) {
  size_t i = (size_t)blockIdx.x * blockDim.x + threadIdx.x;
  size_t stride = (size_t)gridDim.x * blockDim.x;
  float4 z = make_float4(0.f, 0.f, 0.f, 0.f);
  for (; i < n4; i += stride) out[i] = z;
}

// ---------------------------------------------------------------------------
// Kernel 2: support = X @ W via V_WMMA_F32_16X16X4_F32, with the block's
// 16x256 X strip staged into LDS by the Tensor Data Mover (one DMA per block,
// TENSORcnt-tracked). TDM LDS padding skews row pitch to 260 dwords so the
// A-fragment ds_load_b64 pattern (16 rows x 4 consecutive dwords) covers all
// 64 LDS banks exactly once.
//
// fp32 WMMA VGPR layouts (ISA 7.12.2):
//   A 16x4 : lane L -> M = L%16, K = 2*(L/16) + {0,1}   (v2f per lane)
//   B 4x16 : lane L -> N = L%16, K = 2*(L/16) + {0,1}   (v2f per lane)
//   D 16x16: VGPR v -> M = v + 8*(L/16), N = L%16       (v8f per lane)
// ---------------------------------------------------------------------------
__global__ __launch_bounds__(256) void gcn_gemm_wmma(
    const float* __restrict__ X, const float* __restrict__ W,
    float* __restrict__ support, int n_rows) {
  __shared__ float xs[16 * LDS_PITCH];

  const int wave = threadIdx.x >> 5;   // 0..7 -> column tile
  const int lane = threadIdx.x & 31;
  const int half = lane >> 4;          // 0 or 1
  const int l16  = lane & 15;
  const int row0 = blockIdx.x * 16;
  const int col0 = wave * 16;

  int rows_here = n_rows - row0;
  if (rows_here > 16) rows_here = 16;

  // --- TDM: DMA X[row0 .. row0+rows_here) x [0,256) into LDS (wave 0 only;
  // TDM ignores EXEC, so gate at wave granularity). OOB rows read as zero.
  if (threadIdx.x < 32) {
    const unsigned long long ga = (unsigned long long)(const void*)(X + (size_t)row0 * IN_DIM);
    u32x4 g0;
    g0.x = 1u;                                     // count=1, user mode
    g0.y = (unsigned)(uintptr_t)xs;                // lds_addr (byte offset, low 32 of flat)
    g0.z = (unsigned)(ga & 0xFFFFFFFFu);           // global_addr[31:0]
    g0.w = (unsigned)((ga >> 32) & 0x01FFFFFFu)    // global_addr[56:32]
         | (2u << 30);                             // type=2 ("image")
    i32x8 g1;
    g1[0] = (int)((2u << 16)      // data_size: 4 bytes
                | (1u << 20)      // pad_enable
                | (7u << 22)      // pad_interval: 256 dwords (one row)
                | (3u << 25));    // pad_amount: 4 dwords
    g1[1] = (int)((256u & 0xFFFFu) << 16);         // tensor_dim0[15:0]=256 | barrier_addr=0
    g1[2] = (int)(((unsigned)rows_here) << 16);    // tensor_dim1[15:0]    | tensor_dim0[31:16]=0
    g1[3] = (int)(256u << 16);                     // tile_dim0=256        | tensor_dim1[31:16]=0
    g1[4] = 16;                                    // tile_dim1=16, tile_dim2=0
    g1[5] = 256;                                   // tensor_dim0_stride[31:0] = 256 elements
    g1[6] = 0;                                     // stride0[47:32]=0 | stride1 lo (unused, 2D)
    g1[7] = 0;
    i32x4 z4 = {0, 0, 0, 0};                       // groups 2/3 unused (2D tensor)
    i32x8 z8 = {0, 0, 0, 0, 0, 0, 0, 0};           // trailing arg of 6-arg form (unused)
    __builtin_amdgcn_tensor_load_to_lds(g0, g1, z4, z4, z8, 0);
    __builtin_amdgcn_s_wait_tensorcnt(0);
  }
  __syncthreads();

  const float* __restrict__ arow = xs + (size_t)l16 * LDS_PITCH;

  v8f acc = {};
  for (int k0 = 0; k0 < IN_DIM; k0 += 4) {
    const int kb = k0 + half * 2;                  // even -> 8B-aligned ds_load_b64
    v2f a = *reinterpret_cast<const v2f*>(arow + kb);
    v2f b;
    b.x = W[(size_t)kb * OUT_DIM + col0 + l16];
    b.y = W[(size_t)(kb + 1) * OUT_DIM + col0 + l16];
    acc = __builtin_amdgcn_wmma_f32_16x16x4_f32(
        /*neg_a=*/false, a, /*neg_b=*/false, b,
        /*c_mod=*/(short)0, acc, /*reuse_a=*/false, /*reuse_b=*/false);
  }

  // Store: VGPR v holds D[M = v + 8*half][N = l16]
  float* __restrict__ ocol = support + (size_t)row0 * OUT_DIM + col0 + l16;
  if (rows_here == 16) {                           // uniform fast path (always, N%16==0)
#pragma unroll
    for (int v = 0; v < 8; ++v)
      ocol[(size_t)(v + 8 * half) * OUT_DIM] = acc[v];
  } else {
#pragma unroll
    for (int v = 0; v < 8; ++v) {
      const int m = v + 8 * half;
      if (m < rows_here) ocol[(size_t)m * OUT_DIM] = acc[v];
    }
  }
}

// ---------------------------------------------------------------------------
// Kernel 3: out[dst] += weight * support[src]  (segment-sum via L2 f32 atomics)
// One wave per edge iteration: 32 lanes x float4 = 128 columns -> 512B
// coalesced gather + 128 global_atomic_add_f32. support/out (51 MB each)
// stay resident in the 192 MB L2, so the random traffic is on-chip.
// ---------------------------------------------------------------------------
__global__ __launch_bounds__(256) void gcn_scatter(
    const float* __restrict__ support, const int* __restrict__ esrc,
    const int* __restrict__ edst, const float* __restrict__ ewgt,
    float* __restrict__ out, int n_edges) {
  const int lane    = threadIdx.x & 31;
  const int wave_id = blockIdx.x * (blockDim.x >> 5) + (threadIdx.x >> 5);
  const int n_waves = gridDim.x * (blockDim.x >> 5);
  const int c = lane * 4;

  for (int e = wave_id; e < n_edges; e += n_waves) {
    const int   s = esrc[e];
    const int   d = edst[e];
    const float w = ewgt[e];
    const float4 v = *reinterpret_cast<const float4*>(support + (size_t)s * OUT_DIM + c);
    float* __restrict__ o = out + (size_t)d * OUT_DIM + c;
    unsafeAtomicAdd(o + 0, w * v.x);   // -> global_atomic_add_f32 (L2 atomic)
    unsafeAtomicAdd(o + 1, w * v.y);
    unsafeAtomicAdd(o + 2, w * v.z);
    unsafeAtomicAdd(o + 3, w * v.w);
  }
}

// ---------------------------------------------------------------------------
// Inputs (setup_inputs order): x[N,256] f32, edge_src[E] i32, edge_dst[E] i32,
// edge_weight[E] f32, W[256,128] f32. Output: out[N,128] f32.
// d_ws holds support[N,128] f32 (51.2 MB).
// ---------------------------------------------------------------------------
extern "C" void kernel_launch(void* const* d_in, const int* in_sizes, int n_in,
                              void* d_out, int out_size, void* d_ws, size_t ws_size,
                              hipStream_t stream) {
  const float* X    = (const float*)d_in[0];
  const int*   esrc = (const int*)  d_in[1];
  const int*   edst = (const int*)  d_in[2];
  const float* ewgt = (const float*)d_in[3];
  const float* W    = (const float*)d_in[4];
  float* out     = (float*)d_out;
  float* support = (float*)d_ws;

  const int n_rows  = in_sizes[0] / IN_DIM;  // 100000
  const int n_edges = in_sizes[1];           // 1600000

  // 1) out = 0
  gcn_zero<<<1024, 256, 0, stream>>>((float4*)out, (size_t)out_size / 4);

  // 2) support = X @ W (WMMA fp32, TDM-staged A)
  gcn_gemm_wmma<<<(n_rows + 15) / 16, 256, 0, stream>>>(X, W, support, n_rows);

  // 3) out[dst] += w * support[src]
  gcn_scatter<<<2048, 256, 0, stream>>>(support, esrc, edst, ewgt, out, n_edges);
}